// NonLocalMeans_53764400611633
// MI455X (gfx1250) — compile-verified
//
#include <hip/hip_runtime.h>
#include <hip/hip_bf16.h>

#define TILE     32
#define SR       5
#define PR       2
#define HALO     (SR + PR)            // 7
#define HB       (HALO - PR)          // 5: D-region -> Y-buffer row bias
#define YDIM     (TILE + 2 * HALO)    // 46
#define YSTR     48                   // padded row stride (floats)
#define DDIM     (TILE + 2 * PR)      // 36
#define DSTR     37                   // odd stride -> conflict-free columns
#define NY       (YDIM * YDIM)        // 2116 halo elements per plane
#define IMG      512
#define NTHREADS 256

#define N1FULL   5                    // 5*256 = 1280 of 1296 step-1 elements
#define N1TAIL   (DDIM * DDIM - N1FULL * NTHREADS)   // 16
#define N2FULL   4                    // 4*256 = 1024 of 1152 step-2 elements
#define N2TAIL   (DDIM * TILE - N2FULL * NTHREADS)   // 128

// ---- CDNA5 async global->LDS copy (gfx1250), tracked by ASYNCcnt ----------
// Per-lane gather: copies 4 bytes from the global address in vaddr to the LDS address in vdst, GV addressing mode.
// The low 32 bits of a generic pointer to __shared__ are the wave-relative
// LDS offset (generic LDS addresses are {shared_aperture, lds_offset32}).
__device__ __forceinline__ void async_ld_f32(unsigned lds_byte_off, const float* gptr) {
  asm volatile("global_load_async_to_lds_b32 %0, %1, off"
               :: "v"(lds_byte_off), "v"((unsigned long long)gptr)
               : "memory");
}
__device__ __forceinline__ void wait_async0() {
  asm volatile("s_wait_asynccnt 0" ::: "memory");
}
__device__ __forceinline__ unsigned lds_off(const void* p) {
  return (unsigned)(unsigned long long)p;
}

__global__ __launch_bounds__(NTHREADS) void nlm_kernel(
    const float* __restrict__ rgb,   // [B,3,512,512]
    const float* __restrict__ hptr,  // [1]
    float* __restrict__ out)         // [B,3,512,512]
{
  __shared__ float sR[YDIM * YSTR];
  __shared__ float sG[YDIM * YSTR];
  __shared__ float sB[YDIM * YSTR];
  __shared__ float sY[YDIM * YSTR];
  __shared__ float sD[DDIM * DSTR];  // (y - y_shift)^2 over tile + patch halo
  __shared__ float sC[TILE * DSTR];  // vertical 5-sums

  const int tx  = threadIdx.x;              // 0..31 (x in tile)
  const int ty  = threadIdx.y;              // 0..7
  const int tid = ty * 32 + tx;             // 0..255
  const int x0  = blockIdx.x * TILE;
  const int y0  = blockIdx.y * TILE;
  const int b   = blockIdx.z;

  const size_t plane = (size_t)IMG * IMG;
  const float* rgbB  = rgb + (size_t)b * 3 * plane;

  // ---- Stage 46x46 RGB halo (circular wrap) via async global->LDS --------
  const unsigned baseR = lds_off(&sR[0]);
  const unsigned baseG = lds_off(&sG[0]);
  const unsigned baseB = lds_off(&sB[0]);
  for (int idx = tid; idx < NY; idx += NTHREADS) {
    const int u = idx % YDIM, v = idx / YDIM;
    const int gx = (x0 + u - HALO + IMG) & (IMG - 1);
    const int gy = (y0 + v - HALO + IMG) & (IMG - 1);
    const size_t goff = (size_t)gy * IMG + gx;
    const unsigned loff = (unsigned)(v * YSTR + u) * 4u;
    async_ld_f32(baseR + loff, rgbB + goff);
    async_ld_f32(baseG + loff, rgbB + plane + goff);
    async_ld_f32(baseB + loff, rgbB + 2 * plane + goff);
  }
  wait_async0();        // this wave's async LDS writes are complete
  __syncthreads();      // all waves' halo data visible

  // ---- Luminance on the halo ---------------------------------------------
  for (int idx = tid; idx < NY; idx += NTHREADS) {
    const int u = idx % YDIM, v = idx / YDIM;
    const int k = v * YSTR + u;
    sY[k] = 0.299f * sR[k] + 0.587f * sG[k] + 0.114f * sB[k];
  }
  __syncthreads();

  // ---- exp constant: w = exp(-dist/h) = exp2(-dist * log2(e)/h) ----------
  const float h0   = hptr[0];
  const float hh   = (h0 > 0.f ? h0 : 0.f) + 1e-6f;
  const float cexp = -1.44269504088896340736f / hh;

  // ---- Hoist index decompositions out of the 121-shift loop --------------
  // Step 1 covers DDIM*DDIM = 1296 = 5*256 + 16 elements: chunks 0..4 are
  // always full; only chunk 5 is predicated (tid < 16).
  int k1y[6], k1d[6];
#pragma unroll
  for (int j = 0; j < 6; ++j) {
    const int idx = tid + NTHREADS * j;
    const int du = idx % DDIM, dv = idx / DDIM;
    k1y[j] = (dv + HB) * YSTR + (du + HB);  // sY index (unshifted)
    k1d[j] = dv * DSTR + du;                // sD index
  }
  const bool tail1 = (tid < N1TAIL);        // 16 threads
  // Step 2 covers DDIM*TILE = 1152 = 4*256 + 128: chunks 0..3 full; chunk 4
  // predicated (tid < 128).
  int k2d[5];
#pragma unroll
  for (int j = 0; j < 5; ++j) {
    const int idx = tid + NTHREADS * j;
    const int du = idx % DDIM, yy = idx / DDIM;
    k2d[j] = yy * DSTR + du;                // sD base read == sC write index
  }
  const bool tail2 = (tid < N2TAIL);        // 128 threads
  // Step 3: per-thread 4-row strip.
  const int ybase = ty * 4;
  int kc0[4], kr0[4];
#pragma unroll
  for (int r = 0; r < 4; ++r) {
    kc0[r] = (ybase + r) * DSTR + tx;                  // sC horizontal base
    kr0[r] = (ybase + r + HALO) * YSTR + (tx + HALO);  // RGB read base (+shift)
  }

  float accR[4] = {0.f, 0.f, 0.f, 0.f};
  float accG[4] = {0.f, 0.f, 0.f, 0.f};
  float accB[4] = {0.f, 0.f, 0.f, 0.f};
  float accW[4] = {0.f, 0.f, 0.f, 0.f};

  // ---- 121 shifts ---------------------------------------------------------
#pragma unroll 1
  for (int oy = -SR; oy <= SR; ++oy) {
#pragma unroll 1
    for (int ox = -SR; ox <= SR; ++ox) {
      const int so = oy * YSTR + ox;

      // Step 1: squared luminance differences over tile + patch halo.
#pragma unroll
      for (int j = 0; j < N1FULL; ++j) {      // unconditional chunks
        const float d = sY[k1y[j]] - sY[k1y[j] + so];
        sD[k1d[j]] = d * d;
      }
      if (tail1) {                            // single predicated tail
        const float d = sY[k1y[5]] - sY[k1y[5] + so];
        sD[k1d[5]] = d * d;
      }
      __syncthreads();

      // Step 2: vertical 5-sums.
#pragma unroll
      for (int j = 0; j < N2FULL; ++j) {      // unconditional chunks
        const int kb = k2d[j];
        sC[kb] = sD[kb] + sD[kb + DSTR] + sD[kb + 2 * DSTR] +
                 sD[kb + 3 * DSTR] + sD[kb + 4 * DSTR];
      }
      if (tail2) {                            // single predicated tail
        const int kb = k2d[4];
        sC[kb] = sD[kb] + sD[kb + DSTR] + sD[kb + 2 * DSTR] +
                 sD[kb + 3 * DSTR] + sD[kb + 4 * DSTR];
      }
      __syncthreads();

      // Step 3: horizontal 5-sum, weight, accumulate (registers only).
#pragma unroll
      for (int r = 0; r < 4; ++r) {
        const int kc = kc0[r];
        const float S = sC[kc] + sC[kc + 1] + sC[kc + 2] + sC[kc + 3] + sC[kc + 4];
        const float w = __builtin_amdgcn_exp2f(__builtin_amdgcn_sqrtf(S) * cexp);
        const int kr = kr0[r] + so;
        accR[r] += sR[kr] * w;
        accG[r] += sG[kr] * w;
        accB[r] += sB[kr] * w;
        accW[r] += w;
      }
      // Hazards: next step-1 sD writes vs this step-2 sD reads are covered by
      // the barrier after step 2; next step-2 sC writes vs this step-3 sC
      // reads are covered by the next iteration's first barrier. Two barriers
      // per shift suffice.
    }
  }

  // ---- Normalize, clip, store --------------------------------------------
  float* outB = out + (size_t)b * 3 * plane;
#pragma unroll
  for (int r = 0; r < 4; ++r) {
    const int gy = y0 + ybase + r;
    const int gx = x0 + tx;
    const size_t o = (size_t)gy * IMG + gx;
    const float inv = 1.0f / accW[r];
    float vr = accR[r] * inv; vr = vr < 0.f ? 0.f : (vr > 1.f ? 1.f : vr);
    float vg = accG[r] * inv; vg = vg < 0.f ? 0.f : (vg > 1.f ? 1.f : vg);
    float vb = accB[r] * inv; vb = vb < 0.f ? 0.f : (vb > 1.f ? 1.f : vb);
    outB[o]             = vr;
    outB[plane + o]     = vg;
    outB[2 * plane + o] = vb;
  }
}

extern "C" void kernel_launch(void* const* d_in, const int* in_sizes, int n_in,
                              void* d_out, int out_size, void* d_ws, size_t ws_size,
                              hipStream_t stream) {
  const float* rgb = (const float*)d_in[0];
  const float* h   = (const float*)d_in[1];
  float* out       = (float*)d_out;

  const int B = in_sizes[0] / (3 * IMG * IMG);  // 16
  dim3 grid(IMG / TILE, IMG / TILE, B);         // 16 x 16 x 16
  dim3 block(32, 8, 1);                         // 256 threads = 8 wave32
  hipLaunchKernelGGL(nlm_kernel, grid, block, 0, stream, rgb, h, out);
}